// cReLU_percent_1769526526671
// MI455X (gfx1250) — compile-verified
//
#include <hip/hip_runtime.h>
#include <stdint.h>

// Problem constants (from reference: x is (64, 128, 56, 56) fp32, PERCENT=0.5)
#define BATCH 64
#define NPER  401408            // 128*56*56
#define KSEL  200704            // ceil(0.5 * NPER)
#define TPB   256               // 8 waves of 32 on CDNA5

#define HIST_WGS 7              // WGs per row for histogram passes; NPER/7 = 57344 = 56*1024
#define AP_WGS   56             // WGs per row for apply pass
#define AP_ITERS 7              // 56 WGs * 7 iters * 1024 floats = 401408 exactly

#define BINS1 2048              // key bits [31:21]
#define BINS2 2048              // key bits [20:10]
#define BINS3 1024              // key bits [ 9: 0]

// ---------------- order-preserving float <-> u32 key ----------------
__device__ __forceinline__ unsigned f2key(unsigned u) {
    return (u & 0x80000000u) ? ~u : (u | 0x80000000u);
}
__device__ __forceinline__ unsigned key2f(unsigned k) {
    return (k & 0x80000000u) ? (k & 0x7FFFFFFFu) : ~k;
}

// ---------------- CDNA5 async global->LDS copy (ASYNCcnt path) ----------------
__device__ __forceinline__ void async_b128_to_lds(unsigned lds_addr, unsigned voff_bytes,
                                                  const float* base) {
    // GVS mode: 64-bit SGPR base + 32-bit per-lane VGPR byte offset; dest is LDS byte addr.
    asm volatile("global_load_async_to_lds_b128 %0, %1, %2"
                 :: "v"(lds_addr), "v"(voff_bytes), "s"(base)
                 : "memory");
}

// ---------------- scratch zeroing ----------------
__global__ void zero_kernel(unsigned* __restrict__ p, int n) {
    int i = blockIdx.x * TPB + threadIdx.x;
    if (i < n) p[i] = 0u;
}

// ---------------- histogram passes (LDS hist + global merge) ----------------
template <int PASS, int BINS>
__global__ void hist_kernel(const float* __restrict__ x,
                            unsigned* __restrict__ gh,
                            const unsigned* __restrict__ selPrev)
{
    __shared__ unsigned lh[BINS];
    const int row = blockIdx.y;
    const int tid = threadIdx.x;
    for (int b = tid; b < BINS; b += TPB) lh[b] = 0u;
    __syncthreads();

    unsigned selp = 0u;
    if (PASS != 1) selp = selPrev[row];

    const float* xrow = x + (size_t)row * NPER;
    const unsigned chunk = NPER / HIST_WGS;          // 57344, divisible by 1024
    const unsigned base  = (unsigned)blockIdx.x * chunk;

    for (unsigned i = base + tid * 4u; i < base + chunk; i += TPB * 4u) {
        const float4 v = *(const float4*)(xrow + i);
        const float fv[4] = {v.x, v.y, v.z, v.w};
#pragma unroll
        for (int c = 0; c < 4; ++c) {
            const unsigned key = f2key(__float_as_uint(fv[c]));
            if (PASS == 1) {
                atomicAdd(&lh[key >> 21], 1u);                      // ds_add_u32
            } else if (PASS == 2) {
                if ((key >> 21) == selp) atomicAdd(&lh[(key >> 10) & (BINS2 - 1)], 1u);
            } else {
                if ((key >> 10) == selp) atomicAdd(&lh[key & (BINS3 - 1)], 1u);
            }
        }
    }
    __syncthreads();
    for (int b = tid; b < BINS; b += TPB) {
        const unsigned c = lh[b];
        if (c) atomicAdd(&gh[(size_t)row * BINS + b], c);           // sparse global merge
    }
}

// ---------------- top-down scan of a row histogram ----------------
template <int BINS>
__global__ void scan_kernel(const unsigned* __restrict__ gh,
                            const unsigned* __restrict__ kinArr, unsigned kinConst,
                            const unsigned* __restrict__ selPrev, unsigned prevShift,
                            unsigned* __restrict__ selOut,
                            unsigned* __restrict__ kOut,
                            float* __restrict__ thrOut)
{
    __shared__ unsigned sh[BINS];
    __shared__ unsigned part[TPB];
    const int row = blockIdx.x;
    const int tid = threadIdx.x;
    for (int b = tid; b < BINS; b += TPB) sh[b] = gh[(size_t)row * BINS + b];
    __syncthreads();

    constexpr int PB = BINS / TPB;
    unsigned ps = 0u;
    for (int j = 0; j < PB; ++j) ps += sh[tid * PB + j];
    part[tid] = ps;
    __syncthreads();

    if (tid == 0) {
        const unsigned kin = kinArr ? kinArr[row] : kinConst;
        unsigned cum = 0u;                     // count of keys strictly above current position
        int cb = TPB - 1;
        for (; cb > 0; --cb) {                 // find 8/4-bin chunk containing rank kin
            if (cum + part[cb] >= kin) break;
            cum += part[cb];
        }
        int b = cb * PB + (PB - 1);
        for (; b > cb * PB; --b) {             // find the exact bin
            if (cum + sh[b] >= kin) break;
            cum += sh[b];
        }
        const unsigned knext = kin - cum;      // residual rank inside bin b (>= 1)
        const unsigned sel = (selPrev ? (selPrev[row] << prevShift) : 0u) | (unsigned)b;
        if (selOut) selOut[row] = sel;
        if (kOut)   kOut[row]   = knext;
        if (thrOut) thrOut[row] = __uint_as_float(key2f(sel));   // exact k-th largest value
    }
}

// ---------------- apply pass: async global->LDS double-buffered stream ----------------
__global__ void apply_kernel(const float* __restrict__ x,
                             float* __restrict__ out,
                             const float* __restrict__ thr)
{
    __shared__ float4 buf[2][TPB];             // 8 KB, per-thread private slots
    const int row = blockIdx.y;
    const int tid = threadIdx.x;
    const float t = thr[row];
    const float* xrow = x   + (size_t)row * NPER;
    float*       orow = out + (size_t)row * NPER;

    const unsigned chunkBase = (unsigned)blockIdx.x * (AP_ITERS * TPB * 4u); // in floats

    unsigned ldsA[2];
    ldsA[0] = (unsigned)(size_t)&buf[0][tid];  // LDS byte address = low 32 bits of flat addr
    ldsA[1] = (unsigned)(size_t)&buf[1][tid];

    // prime the pipeline
    async_b128_to_lds(ldsA[0], (chunkBase + tid * 4u) * 4u, xrow);

#pragma unroll 1
    for (int i = 0; i < AP_ITERS; ++i) {
        if (i + 1 < AP_ITERS) {
            async_b128_to_lds(ldsA[(i + 1) & 1],
                              (chunkBase + (unsigned)(i + 1) * (TPB * 4u) + tid * 4u) * 4u,
                              xrow);
            asm volatile("s_wait_asynccnt 0x1" ::: "memory");  // oldest copy (tile i) done
        } else {
            asm volatile("s_wait_asynccnt 0x0" ::: "memory");
        }
        const float4 v = buf[i & 1][tid];      // each lane reads the slot it filled: no barrier
        float4 r;
        r.x = (v.x >= t) ? v.x : 0.0f;
        r.y = (v.y >= t) ? v.y : 0.0f;
        r.z = (v.z >= t) ? v.z : 0.0f;
        r.w = (v.w >= t) ? v.w : 0.0f;
        *(float4*)(orow + chunkBase + (unsigned)i * (TPB * 4u) + tid * 4u) = r;
    }
}

// ---------------- launch ----------------
extern "C" void kernel_launch(void* const* d_in, const int* in_sizes, int n_in,
                              void* d_out, int out_size, void* d_ws, size_t ws_size,
                              hipStream_t stream)
{
    (void)in_sizes; (void)n_in; (void)out_size; (void)ws_size;
    const float* x = (const float*)d_in[0];
    float* out = (float*)d_out;

    // Workspace layout (~1.31 MB total)
    unsigned* histA = (unsigned*)d_ws;                       // 64*2048
    unsigned* histB = histA + (size_t)BATCH * BINS1;         // 64*2048
    unsigned* histC = histB + (size_t)BATCH * BINS2;         // 64*1024
    unsigned* sel1  = histC + (size_t)BATCH * BINS3;         // 64
    unsigned* kA    = sel1  + BATCH;                         // 64
    unsigned* sel12 = kA    + BATCH;                         // 64
    unsigned* kB    = sel12 + BATCH;                         // 64
    float*    thr   = (float*)(kB + BATCH);                  // 64

    const int zn = BATCH * (BINS1 + BINS2 + BINS3);          // 327680 words
    zero_kernel<<<(zn + TPB - 1) / TPB, TPB, 0, stream>>>(histA, zn);

    // Radix-select: 11 + 11 + 10 bits -> exact key of the k-th largest per row.
    hist_kernel<1, BINS1><<<dim3(HIST_WGS, BATCH), TPB, 0, stream>>>(x, histA, nullptr);
    scan_kernel<BINS1><<<BATCH, TPB, 0, stream>>>(histA, nullptr, KSEL, nullptr, 0,
                                                  sel1, kA, nullptr);
    hist_kernel<2, BINS2><<<dim3(HIST_WGS, BATCH), TPB, 0, stream>>>(x, histB, sel1);
    scan_kernel<BINS2><<<BATCH, TPB, 0, stream>>>(histB, kA, 0, sel1, 11,
                                                  sel12, kB, nullptr);
    hist_kernel<3, BINS3><<<dim3(HIST_WGS, BATCH), TPB, 0, stream>>>(x, histC, sel12);
    scan_kernel<BINS3><<<BATCH, TPB, 0, stream>>>(histC, kB, 0, sel12, 10,
                                                  nullptr, nullptr, thr);

    // Streaming threshold pass via the CDNA5 async copy engine.
    apply_kernel<<<dim3(AP_WGS, BATCH), TPB, 0, stream>>>(x, out, thr);
}